// BRF_cell_19610820673652
// MI455X (gfx1250) — compile-verified
//
#include <hip/hip_runtime.h>
#include <hip/hip_bf16.h>

// ---------------------------------------------------------------------------
// BRF network: 500 sequential steps of W@u (4096x4096 f32) + elementwise update.
// W (64MB) is L2-resident on MI455X (192MB L2). One kernel per step; matvec via
// V_WMMA_F32_16X16X4_F32 (A = broadcast u chunk, B = 16x4 W tile, D replicates
// the per-row dot products so extraction is a single lane read).
// ---------------------------------------------------------------------------

#define N_CELLS 4096
#define N_STEPS 500
#define DT      0.01f
#define I_AMP   10.0f
#define Q_DECAY 0.95f
#define THETA   1.0f
#define GC      0.1f

// d_out flat layout (floats): time | u^T | v^T | q^T | x | z^T
#define OFF_TIME 0
#define OFF_U    500
#define OFF_V    (500 + N_CELLS * N_STEPS)
#define OFF_Q    (500 + 2 * N_CELLS * N_STEPS)
#define OFF_X    (500 + 3 * N_CELLS * N_STEPS)
#define OFF_Z    (1000 + 3 * N_CELLS * N_STEPS)

// workspace layout (floats): u0 | u1 | v | q | row_sum | pb   (96KB total)
#define WS_U0 0
#define WS_U1 4096
#define WS_V  8192
#define WS_Q  12288
#define WS_RS 16384
#define WS_PB 20480
#define WS_TOTAL 24576

typedef float v2f __attribute__((ext_vector_type(2)));
typedef float v8f __attribute__((ext_vector_type(8)));

// ---------------------------------------------------------------------------
// Zero the state, write the time / input-current output vectors.
// ---------------------------------------------------------------------------
__global__ __launch_bounds__(256) void brf_init_kernel(float* __restrict__ ws,
                                                       float* __restrict__ out) {
    const int i = blockIdx.x * blockDim.x + threadIdx.x;
    if (i < WS_TOTAL) ws[i] = 0.0f;                    // u0,u1,v,q (+rs,pb slots)
    if (i < N_STEPS) {
        out[OFF_TIME + i] = (float)i * DT;
        out[OFF_X + i]    = (i < 2) ? I_AMP : 0.0f;    // pulse for idx in [0,2)
    }
}

// ---------------------------------------------------------------------------
// Precompute pb[c] = p_omega(omega[c]) - b[c]; hoists the sqrt chain out of
// the 500 per-step kernels (per-step b_eff = pb - q is then a single sub).
// ---------------------------------------------------------------------------
__global__ __launch_bounds__(256) void brf_pb_kernel(const float* __restrict__ b,
                                                     const float* __restrict__ omega,
                                                     float* __restrict__ pb) {
    const int c = blockIdx.x * blockDim.x + threadIdx.x;
    if (c < N_CELLS) {
        const float dom  = DT * omega[c];
        const float p_om = (-1.0f + sqrtf(1.0f - dom * dom)) / DT;
        pb[c] = p_om - b[c];
    }
}

// ---------------------------------------------------------------------------
// row_sum[r] = sum_j W[r][j].  One wave per row (8 rows / 256-thread block).
// Also warms L2 with all of W before the step loop.
// ---------------------------------------------------------------------------
__global__ __launch_bounds__(256) void brf_rowsum_kernel(const float* __restrict__ W,
                                                         float* __restrict__ row_sum) {
    const int wave = threadIdx.x >> 5;
    const int lane = threadIdx.x & 31;
    const int row  = blockIdx.x * 8 + wave;
    const float4* wr = (const float4*)(W + (size_t)row * N_CELLS);
    float s = 0.0f;
    for (int i = lane; i < N_CELLS / 4; i += 32) {
        float4 t = wr[i];
        s += t.x + t.y + t.z + t.w;
    }
    #pragma unroll
    for (int off = 16; off > 0; off >>= 1) s += __shfl_down(s, off, 32);
    if (lane == 0) row_sum[row] = s;
}

// ---------------------------------------------------------------------------
// One BRF step. Block = 256 threads = 8 waves, owns 16 rows (cells).
// Wave w handles K-chunk [w*512, w*512+512) of the dot product with 128
// V_WMMA_F32_16X16X4_F32 ops, partials reduced through LDS, then 16 threads
// apply the BRF update and write outputs for time t.
// ---------------------------------------------------------------------------
__global__ __launch_bounds__(256) void brf_step_kernel(
    const float* __restrict__ W, const float* __restrict__ row_sum,
    const float* __restrict__ pb, const float* __restrict__ omega,
    const float* __restrict__ u_in, float* __restrict__ u_out,
    float* __restrict__ v_st, float* __restrict__ q_st,
    float* __restrict__ out, int t) {
    __shared__ float s_u[N_CELLS];
    __shared__ float s_part[8 * 16];

    const int tid = threadIdx.x;

    // Stage full u vector into LDS (coalesced float4 copies).
    {
        const float4* src = (const float4*)u_in;
        float4*       dst = (float4*)s_u;
        #pragma unroll
        for (int i = 0; i < 4; ++i) dst[tid + 256 * i] = src[tid + 256 * i];
    }
    __syncthreads();

    const int wave  = tid >> 5;
    const int lane  = tid & 31;
    const int row0  = blockIdx.x * 16;
    const int kbase = wave * 512;
    const int half  = (lane < 16) ? 0 : 2;   // K-pair selector per ISA layout
    const int r     = lane & 15;             // row within the 16-row tile

    // B-matrix tile pointer: lane holds W[row0+r][k .. k+1] (or k+2..k+3).
    const float* wp = W + (size_t)(row0 + r) * N_CELLS + kbase + half;
    // A-matrix (broadcast u) pointer in LDS.
    const float* up = s_u + kbase + half;

    v8f acc = {};
    #pragma unroll 8
    for (int k = 0; k < 128; ++k) {
        v2f bmat = *(const v2f*)wp;  wp += 4;   // 16x4 W tile (B operand)
        v2f amat = *(const v2f*)up;  up += 4;   // u[k..k+3] broadcast (A operand)
        // D[m][n] = sum_k u[k] * W[row0+n][k]  (replicated over m)
        acc = __builtin_amdgcn_wmma_f32_16x16x4_f32(
            false, amat, false, bmat, (short)0, acc, false, false);
    }

    // D VGPR0: lanes 0-15 hold D[0][lane] = y_partial[row0+lane].
    if (lane < 16) s_part[wave * 16 + r] = acc[0];
    __syncthreads();

    if (tid < 16) {
        float y = 0.0f;
        #pragma unroll
        for (int w = 0; w < 8; ++w) y += s_part[w * 16 + tid];

        const int cell = row0 + tid;
        const float u  = u_in[cell];
        const float vv = v_st[cell];
        const float qq = q_st[cell];
        const float om = omega[cell];

        const float i_gap = GC * (row_sum[cell] * u - y);   // mask == 1 (G_SPARSE=0)
        const float x_t   = (t < 2) ? I_AMP : 0.0f;
        const float b_eff = pb[cell] - qq;                  // p_omega - b - q

        const float un = u  + DT * (u * b_eff - vv * om + x_t + i_gap);
        const float vn = vv + DT * (u * om + vv * b_eff);
        const float z  = (un - THETA - qq > 0.0f) ? 1.0f : 0.0f;
        const float qn = Q_DECAY * qq + z;

        u_out[cell] = un;
        v_st[cell]  = vn;
        q_st[cell]  = qn;

        const size_t o = (size_t)cell * N_STEPS + t;
        out[OFF_U + o] = un;
        out[OFF_V + o] = vn;
        out[OFF_Q + o] = qn;
        out[OFF_Z + o] = z;
    }
}

// ---------------------------------------------------------------------------
extern "C" void kernel_launch(void* const* d_in, const int* in_sizes, int n_in,
                              void* d_out, int out_size, void* d_ws, size_t ws_size,
                              hipStream_t stream) {
    const float* W     = (const float*)d_in[0];
    const float* b     = (const float*)d_in[1];
    const float* omega = (const float*)d_in[2];
    float* out = (float*)d_out;
    float* ws  = (float*)d_ws;

    float* u0 = ws + WS_U0;
    float* u1 = ws + WS_U1;
    float* vv = ws + WS_V;
    float* qq = ws + WS_Q;
    float* rs = ws + WS_RS;
    float* pbv = ws + WS_PB;

    brf_init_kernel<<<(WS_TOTAL + 255) / 256, 256, 0, stream>>>(ws, out);
    brf_pb_kernel<<<N_CELLS / 256, 256, 0, stream>>>(b, omega, pbv);
    brf_rowsum_kernel<<<N_CELLS / 8, 256, 0, stream>>>(W, rs);

    for (int t = 0; t < N_STEPS; ++t) {
        float* uin  = (t & 1) ? u1 : u0;
        float* uout = (t & 1) ? u0 : u1;
        brf_step_kernel<<<N_CELLS / 16, 256, 0, stream>>>(
            W, rs, pbv, omega, uin, uout, vv, qq, out, t);
    }
}